// MyRNN_90228672954510
// MI455X (gfx1250) — compile-verified
//
#include <hip/hip_runtime.h>
#include <math.h>

// Problem constants (from reference)
#define BATCH 512
#define SEQT  512
#define EMB   100
#define UNITS 64
#define KX    (EMB / 4)    // 25 K-chunks for x @ Wxh
#define KH    (UNITS / 4)  // 16 K-chunks for h @ Whh

typedef __attribute__((ext_vector_type(2))) float v2f;
typedef __attribute__((ext_vector_type(8))) float v8f;

// h tile rows padded to 66 floats: stride 66 mod 64 banks = 2 -> the 16 rows
// read by lanes 0..15 land on distinct banks; 66 is even so v2f loads stay
// 8B-aligned.
#define H_S 66

__global__ __launch_bounds__(128) void rnn_fused_wmma(
    const int*   __restrict__ toks,   // [BATCH, SEQT] int32
    const float* __restrict__ emb,    // [1000, EMB]
    const float* __restrict__ Wxh,    // [EMB, UNITS]
    const float* __restrict__ Whh,    // [UNITS, UNITS]
    const float* __restrict__ bias,   // [UNITS]
    const float* __restrict__ Wout,   // [UNITS]
    const float* __restrict__ bout,   // [1]
    float*       __restrict__ out)    // [BATCH]
{
  __shared__ float sH[2][16 * H_S];   // double-buffered h tile: 16 rows x 64 units

  const int tid  = threadIdx.x;
  const int wave = tid >> 5;          // 4 wave32 waves per block
  const int lane = tid & 31;
  const int m    = lane & 15;         // A-matrix row / D-matrix column index
  const int half = lane >> 4;         // 0: K=4k+{0,1}, 1: K=4k+{2,3}
  const int n0   = wave * 16;         // this wave's 16-unit N-tile
  const int b0   = blockIdx.x * 16;   // this block's 16 batch rows

  // ---- Hoist loop-invariant B-matrix fragments into registers (once). ----
  // B layout (16x16x4 f32): VGPR0 = row K=4k+2*half, VGPR1 = row K=4k+2*half+1,
  // column N = n0 + m for this lane.
  v2f bx[KX];
#pragma unroll
  for (int kc = 0; kc < KX; ++kc) {
    const int k = kc * 4 + half * 2;
    bx[kc].x = Wxh[(k + 0) * UNITS + n0 + m];
    bx[kc].y = Wxh[(k + 1) * UNITS + n0 + m];
  }
  v2f bh[KH];
#pragma unroll
  for (int kc = 0; kc < KH; ++kc) {
    const int k = kc * 4 + half * 2;
    bh[kc].x = Whh[(k + 0) * UNITS + n0 + m];
    bh[kc].y = Whh[(k + 1) * UNITS + n0 + m];
  }
  const float bv = bias[n0 + m];      // bias depends only on N

  // h0 = 0
  for (int i = tid; i < 16 * H_S; i += 128) sH[0][i] = 0.0f;
  __syncthreads();

  // Fused embedding gather + x_t @ Wxh (+ bias) into a fresh accumulator.
  // Independent of the recurrent state -> pipelined one step ahead.
  auto compute_x = [&](int t) -> v8f {
    v8f x = {bv, bv, bv, bv, bv, bv, bv, bv};
    const int   tok  = toks[(b0 + m) * SEQT + t];
    const float* erow = emb + tok * EMB;
#pragma unroll
    for (int kc = 0; kc < KX; ++kc) {
      const int k = kc * 4 + half * 2;
      v2f a = *(const v2f*)(erow + k);   // A[m][k], A[m][k+1] (8B aligned)
      x = __builtin_amdgcn_wmma_f32_16x16x4_f32(
          false, a, false, bx[kc], (short)0, x, false, false);
    }
    return x;
  };

  v8f xacc = compute_x(0);

  // ---- Sequential scan over time (software-pipelined) ----
  for (int t = 0; t < SEQT; ++t) {
    // Step t+1's x-projection: independent WMMA chain + global loads that the
    // scheduler interleaves with the dependent h-chain below.
    const int tn   = (t + 1 < SEQT) ? (t + 1) : t;   // last iter: redundant, discarded
    v8f       xnxt = compute_x(tn);

    // h_t @ Whh accumulated onto xacc (dependent chain of 16 WMMAs).
    v8f acc = xacc;
    const float* hin = sH[t & 1];
#pragma unroll
    for (int kc = 0; kc < KH; ++kc) {
      const int k = kc * 4 + half * 2;
      v2f a = *(const v2f*)(hin + m * H_S + k);
      acc = __builtin_amdgcn_wmma_f32_16x16x4_f32(
          false, a, false, bh[kc], (short)0, acc, false, false);
    }

    // tanh + write h_{t+1}. D layout: VGPR r -> row M = r + 8*half, col N = n0+m.
    float* hout = sH[(t + 1) & 1];
#pragma unroll
    for (int r = 0; r < 8; ++r) {
      hout[(r + half * 8) * H_S + n0 + m] = tanhf(acc[r]);
    }
    __syncthreads();   // one barrier/step: double-buffering covers both hazards
    xacc = xnxt;
  }

  // ---- Output head: sigmoid(h_T @ Wout + bout). Final h is in sH[SEQT & 1] = sH[0].
  if (tid < 16) {
    const float* hf = sH[0] + tid * H_S;
    float s = 0.0f;
#pragma unroll
    for (int u = 0; u < UNITS; ++u) s += hf[u] * Wout[u];
    s += bout[0];
    out[b0 + tid] = 1.0f / (1.0f + expf(-s));
  }
}

extern "C" void kernel_launch(void* const* d_in, const int* in_sizes, int n_in,
                              void* d_out, int out_size, void* d_ws, size_t ws_size,
                              hipStream_t stream) {
  (void)in_sizes; (void)n_in; (void)d_ws; (void)ws_size; (void)out_size;
  const int*   toks = (const int*)  d_in[0];
  const float* emb  = (const float*)d_in[1];
  const float* Wxh  = (const float*)d_in[2];
  const float* Whh  = (const float*)d_in[3];
  const float* bias = (const float*)d_in[4];
  const float* Wout = (const float*)d_in[5];
  const float* bout = (const float*)d_in[6];
  float* out = (float*)d_out;

  dim3 grid(BATCH / 16);   // 32 blocks, each owns 16 batch rows for all 512 steps
  dim3 block(128);         // 4 wave32 waves: one 16-unit N-tile each
  hipLaunchKernelGGL(rnn_fused_wmma, grid, block, 0, stream,
                     toks, emb, Wxh, Whh, bias, Wout, bout, out);
}